// PowerSpectrum_67345087201458
// MI455X (gfx1250) — compile-verified
//
#include <hip/hip_runtime.h>
#include <hip/hip_bf16.h>
#include <math.h>

#define ROI   512
#define N2    (ROI * ROI)        // 262144
#define NB    128                // batch
#define NFREQ 257                // roi/2 + 1
#define PI_F  3.14159265358979323846f
#define TWO_PI_F 6.28318530717958647692f

typedef __attribute__((ext_vector_type(2))) float v2f;
typedef __attribute__((ext_vector_type(8))) float v8f;

// ---------------------------------------------------------------------------
// WMMA helpers: D(16x16,f32) = A(16x4,f32) * B(4x16,f32) + C
// A layout: lane<16 -> row M=lane, holds K=0(x),K=1(y); lane>=16 -> K=2,K=3
// B layout: lane<16 -> col N=lane, holds K=0(x),K=1(y); lane>=16 -> K=2,K=3
// D layout: vgpr i: lanes0-15 -> M=i, lanes16-31 -> M=i+8; N = lane&15
// ---------------------------------------------------------------------------
__device__ __forceinline__ v8f wmma4(v2f a, v2f b, v8f c) {
  return __builtin_amdgcn_wmma_f32_16x16x4_f32(false, a, false, b,
                                               (short)0, c, false, false);
}

__device__ __forceinline__ v2f load_afrag(const float* __restrict__ A,
                                          int rowBase, int kk, int lane) {
  int half = lane >> 4, m = lane & 15;
  const float* p = A + (size_t)(rowBase + m) * ROI + kk + 2 * half;
  v2f a; a.x = p[0]; a.y = p[1];  // 8B contiguous -> global_load_b64
  return a;
}

__device__ __forceinline__ v2f load_bfrag(const float* __restrict__ B,
                                          int colBase, int kk, int lane) {
  int half = lane >> 4, n = lane & 15;
  int k0 = kk + 2 * half;
  v2f b;
  b.x = B[(size_t)k0 * ROI + colBase + n];
  b.y = B[(size_t)(k0 + 1) * ROI + colBase + n];
  return b;
}

// ---------------------------------------------------------------------------
// precompute: hann window, DFT cos/sin matrices, zero accumulators
// ---------------------------------------------------------------------------
__global__ void ps_precompute_k(float* __restrict__ hann,
                                float* __restrict__ Cr,
                                float* __restrict__ Ci,
                                float* __restrict__ ps,
                                float* __restrict__ sums,
                                float* __restrict__ counts) {
  int tid = blockIdx.x * blockDim.x + threadIdx.x;
  if (tid >= N2) return;
  int i = tid >> 9, j = tid & (ROI - 1);

  // linspace(-256, 256, 512): step = 512/511
  float li = -256.0f + (float)i * (512.0f / 511.0f);
  float lj = -256.0f + (float)j * (512.0f / 511.0f);
  float r = sqrtf(li * li + lj * lj);
  float h = 0.5f * (1.0f + cosf(PI_F * r / 256.0f));
  hann[tid] = (r > 256.0f) ? 0.0f : h;

  // W[j,k] = exp(-2*pi*i*jk/512); periodic -> reduce jk mod 512 (fp32-safe)
  int t = (i * j) & (ROI - 1);
  float theta = TWO_PI_F * (float)t / (float)ROI;
  float s, c;
  sincosf(theta, &s, &c);
  Cr[tid] = c;
  Ci[tid] = -s;

  ps[tid] = 0.0f;
  if (tid < NFREQ) { sums[tid] = 0.0f; counts[tid] = 0.0f; }
}

// ---------------------------------------------------------------------------
// per-image mean (LDS tree reduction)
// ---------------------------------------------------------------------------
__global__ void ps_mean_k(const float* __restrict__ batch,
                          float* __restrict__ mean) {
  __shared__ float red[256];
  int b = blockIdx.x;
  const float* img = batch + (size_t)b * N2;
  float s = 0.0f;
  for (int t = threadIdx.x; t < N2; t += 256) s += img[t];
  red[threadIdx.x] = s;
  __syncthreads();
  for (int off = 128; off > 0; off >>= 1) {
    if (threadIdx.x < off) red[threadIdx.x] += red[threadIdx.x + off];
    __syncthreads();
  }
  if (threadIdx.x == 0) mean[b] = red[0] * (1.0f / (float)N2);
}

// ---------------------------------------------------------------------------
// windowed, mean-subtracted chunk: Xw[b] = (batch[b] - mean[b]) * hann
// ---------------------------------------------------------------------------
__global__ void ps_window_k(const float* __restrict__ batch,
                            const float* __restrict__ mean,
                            const float* __restrict__ hann,
                            float* __restrict__ Xw, int nimg) {
  int idx = blockIdx.x * blockDim.x + threadIdx.x;
  if (idx >= nimg * N2) return;
  int b = idx >> 18;           // N2 == 1<<18
  int r = idx & (N2 - 1);
  Xw[idx] = (batch[idx] - mean[b]) * hann[r];
}

// ---------------------------------------------------------------------------
// stage 1: Yr = Cr * Xw, Yi = Ci * Xw   (one 16x16 tile per wave, per image)
// ---------------------------------------------------------------------------
__global__ void __launch_bounds__(32)
ps_stage1_k(const float* __restrict__ Cr, const float* __restrict__ Ci,
            const float* __restrict__ Xw,
            float* __restrict__ Yr, float* __restrict__ Yi) {
  int lane = threadIdx.x & 31;
  int rowBase = blockIdx.y * 16;
  int colBase = blockIdx.x * 16;
  size_t ioff = (size_t)blockIdx.z * N2;
  const float* B = Xw + ioff;

  v8f accR = {}, accI = {};
#pragma unroll 4
  for (int kk = 0; kk < ROI; kk += 4) {
    v2f bf = load_bfrag(B, colBase, kk, lane);
    v2f ar = load_afrag(Cr, rowBase, kk, lane);
    v2f ai = load_afrag(Ci, rowBase, kk, lane);
    accR = wmma4(ar, bf, accR);
    accI = wmma4(ai, bf, accI);
  }

  int half = lane >> 4, m = lane & 15;
  float* yr = Yr + ioff;
  float* yi = Yi + ioff;
#pragma unroll
  for (int i = 0; i < 8; ++i) {
    int row = rowBase + half * 8 + i;
    yr[(size_t)row * ROI + colBase + m] = accR[i];
    yi[(size_t)row * ROI + colBase + m] = accI[i];
  }
}

// ---------------------------------------------------------------------------
// stage 2: Zr = Yr*Cr - Yi*Ci, Zi = Yr*Ci + Yi*Cr ; ps += scale*(Zr^2+Zi^2)
// ---------------------------------------------------------------------------
__global__ void __launch_bounds__(32)
ps_stage2_k(const float* __restrict__ Cr, const float* __restrict__ Ci,
            const float* __restrict__ Yr, const float* __restrict__ Yi,
            float* __restrict__ ps) {
  int lane = threadIdx.x & 31;
  int rowBase = blockIdx.y * 16;
  int colBase = blockIdx.x * 16;
  size_t ioff = (size_t)blockIdx.z * N2;
  const float* Ar = Yr + ioff;
  const float* Ai = Yi + ioff;

  v8f a_rr = {}, a_ii = {}, a_ri = {}, a_ir = {};
#pragma unroll 2
  for (int kk = 0; kk < ROI; kk += 4) {
    v2f ar = load_afrag(Ar, rowBase, kk, lane);
    v2f ai = load_afrag(Ai, rowBase, kk, lane);
    v2f br = load_bfrag(Cr, colBase, kk, lane);
    v2f bi = load_bfrag(Ci, colBase, kk, lane);
    a_rr = wmma4(ar, br, a_rr);
    a_ii = wmma4(ai, bi, a_ii);
    a_ri = wmma4(ar, bi, a_ri);
    a_ir = wmma4(ai, br, a_ir);
  }

  // nps2D scale: px^2/roi^2, and mean over 128 images
  const float SCALE = (float)(0.01 / ((double)N2 * (double)NB));
  int half = lane >> 4, m = lane & 15;
#pragma unroll
  for (int i = 0; i < 8; ++i) {
    float zr = a_rr[i] - a_ii[i];
    float zi = a_ri[i] + a_ir[i];
    int row = rowBase + half * 8 + i;
    atomicAdd(ps + (size_t)row * ROI + colBase + m,
              SCALE * (zr * zr + zi * zi));
  }
}

// ---------------------------------------------------------------------------
// fftshift write-out of nps2D + radial binning atomics
// ---------------------------------------------------------------------------
__global__ void ps_bin_k(const float* __restrict__ ps,
                         float* __restrict__ sums,
                         float* __restrict__ counts,
                         float* __restrict__ out) {
  int tid = blockIdx.x * blockDim.x + threadIdx.x;
  if (tid >= N2) return;
  int i = tid >> 9, j = tid & (ROI - 1);
  // fftshift: dst(i,j) <- src((i+256)&511, (j+256)&511)
  int si = (i + 256) & (ROI - 1);
  int sj = (j + 256) & (ROI - 1);
  float v = ps[(size_t)si * ROI + sj];
  out[NFREQ + tid] = v;  // nps2D region of output

  int dx = i - 256, dy = j - 256;
  int rad = (int)rintf(sqrtf((float)(dx * dx + dy * dy)));  // half-even
  if (rad < NFREQ) {
    atomicAdd(sums + rad, v);
    atomicAdd(counts + rad, 1.0f);
  }
}

__global__ void ps_fin_k(const float* __restrict__ sums,
                         const float* __restrict__ counts,
                         float* __restrict__ out) {
  int tid = blockIdx.x * blockDim.x + threadIdx.x;
  if (tid >= NFREQ) return;
  out[tid] = sums[tid] / counts[tid];                       // nps1D
  out[NFREQ + N2 + tid] = 5.0f * (float)tid / 256.0f;       // f1D (nyquist=5)
}

// ---------------------------------------------------------------------------
extern "C" void kernel_launch(void* const* d_in, const int* in_sizes, int n_in,
                              void* d_out, int out_size, void* d_ws,
                              size_t ws_size, hipStream_t stream) {
  (void)in_sizes; (void)n_in; (void)out_size;
  const float* batch = (const float*)d_in[0];
  float* out = (float*)d_out;
  float* ws = (float*)d_ws;

  // workspace layout (floats)
  float* hann   = ws;
  float* Cr     = ws + (size_t)N2;
  float* Ci     = ws + (size_t)2 * N2;
  float* ps     = ws + (size_t)3 * N2;
  float* mean   = ws + (size_t)4 * N2;
  float* sums   = mean + 128;
  float* counts = sums + NFREQ;
  const size_t fixed_floats = (size_t)4 * N2 + 1024;
  float* chunk = ws + fixed_floats;

  // chunk size from ws_size (deterministic across calls)
  long avail = (long)(ws_size / sizeof(float)) - (long)fixed_floats;
  int CH = 1;
  if (avail > 0) {
    long c = avail / (3L * N2);
    CH = (c < 1) ? 1 : (c > NB ? NB : (int)c);
  }
  float* Xw = chunk;
  float* Yr = Xw + (size_t)CH * N2;
  float* Yi = Yr + (size_t)CH * N2;

  ps_precompute_k<<<N2 / 256, 256, 0, stream>>>(hann, Cr, Ci, ps, sums, counts);
  ps_mean_k<<<NB, 256, 0, stream>>>(batch, mean);

  for (int b0 = 0; b0 < NB; b0 += CH) {
    int ch = (NB - b0) < CH ? (NB - b0) : CH;
    ps_window_k<<<(ch * N2 + 255) / 256, 256, 0, stream>>>(
        batch + (size_t)b0 * N2, mean + b0, hann, Xw, ch);
    dim3 g(ROI / 16, ROI / 16, ch);
    ps_stage1_k<<<g, 32, 0, stream>>>(Cr, Ci, Xw, Yr, Yi);
    ps_stage2_k<<<g, 32, 0, stream>>>(Cr, Ci, Yr, Yi, ps);
  }

  ps_bin_k<<<N2 / 256, 256, 0, stream>>>(ps, sums, counts, out);
  ps_fin_k<<<2, 256, 0, stream>>>(sums, counts, out);
}